// CARAFEPlusPlusDownsample_78348793414243
// MI455X (gfx1250) — compile-verified
//
#include <hip/hip_runtime.h>

typedef __attribute__((ext_vector_type(2))) float v2f;
typedef __attribute__((ext_vector_type(8))) float v8f;

#define W_IN   160
#define HW_IN  25600   // 160*160
#define W_OUT  80
#define HW_OUT 6400    // 80*80
#define C_IN   128
#define C_MID  16
#define K_ENC  25
#define NB     8

// ---------------------------------------------------------------------------
// Kernel A: compressed[b,m,p] = sum_k w[m,k] * x[b,k,p]
// GEMM M=16, K=128, N=204800 via V_WMMA_F32_16X16X4_F32.
// One wave computes a 16-outch x 16-pixel tile; 32 WMMAs over K=128.
// f32 A/B layout: lanes 0-15 hold K={0,1} in VGPR0/1, lanes 16-31 hold K={2,3}.
// ---------------------------------------------------------------------------
__global__ __launch_bounds__(256) void carafe_compress_wmma(
    const float* __restrict__ x, const float* __restrict__ w,
    float* __restrict__ comp)
{
    __shared__ float sw[C_MID * C_IN]; // 16x128 weights, 8 KB
    for (int i = threadIdx.x; i < C_MID * C_IN; i += 256) sw[i] = w[i];
    __syncthreads();

    const int lane = threadIdx.x & 31;
    const int wave = threadIdx.x >> 5;
    const int tile = blockIdx.x * 8 + wave;   // 16 pixels per tile
    const int pg   = tile * 16;
    const int b    = pg / HW_IN;
    const int p    = pg - b * HW_IN;
    const int n    = lane & 15;               // pixel column within tile
    const int kh   = (lane >> 4) * 2;         // K sub-offset per half-wave

    const float* xb = x + (size_t)b * C_IN * HW_IN + p + n;
    const float* wl = sw + (lane & 15) * C_IN;  // A row m = lane&15

    v8f acc = {0.f, 0.f, 0.f, 0.f, 0.f, 0.f, 0.f, 0.f};
    #pragma unroll 4
    for (int kk = 0; kk < C_IN; kk += 4) {
        const int k0 = kk + kh;
        v2f a  = { wl[k0], wl[k0 + 1] };
        v2f bv = { xb[(size_t)k0 * HW_IN], xb[(size_t)(k0 + 1) * HW_IN] };
        acc = __builtin_amdgcn_wmma_f32_16x16x4_f32(
                  /*neg_a=*/false, a, /*neg_b=*/false, bv,
                  /*c_mod=*/(short)0, acc, /*reuse_a=*/false, /*reuse_b=*/false);
    }

    float* cb = comp + (size_t)b * C_MID * HW_IN + p + n;
    const int mh = (lane >> 4) * 8;           // C/D: M = v + 8*(lane>=16)
    #pragma unroll
    for (int v = 0; v < 8; ++v)
        cb[(size_t)(v + mh) * HW_IN] = acc[v];
}

// ---------------------------------------------------------------------------
// Kernel B: 3x3 stride-2 pad-1 conv (16ch -> 25ch) + softmax over the 25.
// One thread per (b, y, x). Tap offsets/masks hoisted out of the ic loop.
// ---------------------------------------------------------------------------
__global__ __launch_bounds__(256) void carafe_encoder_softmax(
    const float* __restrict__ comp, const float* __restrict__ we,
    float* __restrict__ kern)
{
    __shared__ float swe[K_ENC * C_MID * 9]; // 3600 floats, 14.4 KB
    for (int i = threadIdx.x; i < K_ENC * C_MID * 9; i += 256) swe[i] = we[i];
    __syncthreads();

    const int t  = blockIdx.x * 256 + threadIdx.x;
    const int b  = t / HW_OUT;
    const int r  = t - b * HW_OUT;
    const int y  = r / W_OUT;
    const int xo = r - y * W_OUT;

    int   off[9];
    float msk[9];
    #pragma unroll
    for (int dy = 0; dy < 3; ++dy)
        #pragma unroll
        for (int dx = 0; dx < 3; ++dx) {
            const int iy = 2 * y + dy - 1, ix = 2 * xo + dx - 1;
            const bool ok = (iy >= 0) & (iy < W_IN) & (ix >= 0) & (ix < W_IN);
            off[dy * 3 + dx] = ok ? (iy * W_IN + ix) : 0;
            msk[dy * 3 + dx] = ok ? 1.f : 0.f;
        }

    float acc[K_ENC];
    #pragma unroll
    for (int o = 0; o < K_ENC; ++o) acc[o] = 0.f;

    const float* cb = comp + (size_t)b * C_MID * HW_IN;
    #pragma unroll 1
    for (int ic = 0; ic < C_MID; ++ic) {
        const float* cc = cb + (size_t)ic * HW_IN;
        float tp[9];
        #pragma unroll
        for (int q = 0; q < 9; ++q) tp[q] = cc[off[q]] * msk[q];
        #pragma unroll
        for (int o = 0; o < K_ENC; ++o) {
            const float* wl = swe + (o * C_MID + ic) * 9;
            float s = acc[o];
            #pragma unroll
            for (int q = 0; q < 9; ++q) s += tp[q] * wl[q];
            acc[o] = s;
        }
    }

    // numerically stable softmax over 25 channels
    float mx = acc[0];
    #pragma unroll
    for (int o = 1; o < K_ENC; ++o) mx = fmaxf(mx, acc[o]);
    float sum = 0.f;
    #pragma unroll
    for (int o = 0; o < K_ENC; ++o) { acc[o] = __expf(acc[o] - mx); sum += acc[o]; }
    const float inv = 1.f / sum;

    float* kb = kern + (size_t)b * K_ENC * HW_OUT + y * W_OUT + xo;
    #pragma unroll
    for (int o = 0; o < K_ENC; ++o) kb[(size_t)o * HW_OUT] = acc[o] * inv;
}

// ---------------------------------------------------------------------------
// Kernel C: content-aware reassembly.
// One thread per (b, y, x) handling all 128 channels; the 25 per-pixel softmax
// weights are loaded once into registers (zeroed for out-of-bounds taps, which
// exactly reproduces the reference's zero padding).
// ---------------------------------------------------------------------------
__global__ __launch_bounds__(256) void carafe_reassemble(
    const float* __restrict__ x, const float* __restrict__ kern,
    float* __restrict__ out)
{
    const int t  = blockIdx.x * 256 + threadIdx.x;
    const int b  = t / HW_OUT;
    const int r  = t - b * HW_OUT;
    const int y  = r / W_OUT;
    const int xo = r - y * W_OUT;

    const float* kb = kern + (size_t)b * K_ENC * HW_OUT + y * W_OUT + xo;
    int   off[K_ENC];
    float kv[K_ENC];
    #pragma unroll
    for (int i = 0; i < 5; ++i)
        #pragma unroll
        for (int j = 0; j < 5; ++j) {
            const int q  = i * 5 + j;
            const int iy = 2 * y + i - 2, ix = 2 * xo + j - 2;
            const bool ok = (iy >= 0) & (iy < W_IN) & (ix >= 0) & (ix < W_IN);
            off[q] = ok ? (iy * W_IN + ix) : 0;
            kv[q]  = ok ? kb[(size_t)q * HW_OUT] : 0.f;
        }

    const float* xb = x + (size_t)b * C_IN * HW_IN;
    float* ob = out + (size_t)b * C_IN * HW_OUT + y * W_OUT + xo;
    #pragma unroll 1
    for (int c = 0; c < C_IN; ++c) {
        const float* xc = xb + (size_t)c * HW_IN;
        float s = 0.f;
        #pragma unroll
        for (int q = 0; q < K_ENC; ++q) s += xc[off[q]] * kv[q];
        ob[(size_t)c * HW_OUT] = s;
    }
}

// ---------------------------------------------------------------------------
extern "C" void kernel_launch(void* const* d_in, const int* in_sizes, int n_in,
                              void* d_out, int out_size, void* d_ws, size_t ws_size,
                              hipStream_t stream)
{
    const float* x  = (const float*)d_in[0];  // (8,128,160,160)
    const float* wc = (const float*)d_in[1];  // (16,128,1,1)
    const float* we = (const float*)d_in[2];  // (25,16,3,3)
    float* out  = (float*)d_out;              // (8,128,80,80)

    float* comp = (float*)d_ws;                              // 8*16*25600 f32 = 13.1 MB
    float* kern = comp + (size_t)NB * C_MID * HW_IN;         // 8*25*6400  f32 =  5.1 MB

    // A: 204800 pixels / (8 waves * 16 px) = 1600 blocks
    carafe_compress_wmma<<<1600, 256, 0, stream>>>(x, wc, comp);
    // B/C: 8*80*80 = 51200 threads = 200 blocks
    carafe_encoder_softmax<<<200, 256, 0, stream>>>(comp, we, kern);
    carafe_reassemble<<<200, 256, 0, stream>>>(x, kern, out);
}